// EdgeAttrGATBlock_38371237822900
// MI455X (gfx1250) — compile-verified
//
#include <hip/hip_runtime.h>
#include <hip/hip_bf16.h>
#include <math.h>

#define DIM      128
#define HEADS    4
#define CH       32
#define N_NODES  50000
#define N_EDGES  800000
#define NEG_SLOPE 0.2f
#define LN_EPS   1e-5f
#define SM_EPS   1e-16f

typedef float v2f __attribute__((ext_vector_type(2)));
typedef float v8f __attribute__((ext_vector_type(8)));

// ---------------- workspace layout (floats) ----------------
// xl      : [N, 128]            @ 0
// alpha   : [E, 4]              @ OFF_ALPHA   (logits, then exp(..))
// amax    : [N, 4]              @ OFF_AMAX
// denom   : [N, 4]              @ OFF_DENOM
// outacc  : [N, 128]            @ OFF_OUT
#define OFF_XL     0
#define OFF_ALPHA  (N_NODES * DIM)                       // 6,400,000
#define OFF_AMAX   (OFF_ALPHA + N_EDGES * HEADS)         // 9,600,000
#define OFF_DENOM  (OFF_AMAX + N_NODES * HEADS)          // 9,800,000
#define OFF_OUT    (OFF_DENOM + N_NODES * HEADS)         // 10,000,000

// ---------------- K0: init accumulators ----------------
__global__ void k0_init(float* __restrict__ ws) {
    int idx = blockIdx.x * blockDim.x + threadIdx.x;   // over N*DIM
    if (idx < N_NODES * DIM) ws[OFF_OUT + idx] = 0.0f;
    if (idx < N_NODES * HEADS) {
        ws[OFF_AMAX + idx]  = -INFINITY;
        ws[OFF_DENOM + idx] = 0.0f;
    }
}

// ---------------- K1: xl = x @ W_l + b_l  (WMMA f32 16x16x4) ----------------
// One wave computes a 16x16 output tile. 8 waves/block cover the 8 column
// tiles of the 128-wide output; blockIdx.x covers 3125 row tiles (50000/16).
__global__ void k1_gemm_xl(const float* __restrict__ x,
                           const float* __restrict__ W,   // [128,128] row-major (in,out)
                           const float* __restrict__ b,   // [128]
                           float* __restrict__ xl) {
    const int wave  = threadIdx.x >> 5;     // 0..7 -> column tile
    const int lane  = threadIdx.x & 31;
    const int l15   = lane & 15;
    const int lhalf = lane >> 4;            // 0 or 1 (K split per ISA A/B layout)
    const int m0    = blockIdx.x * 16;
    const int n0    = wave * 16;

    v8f c = {0.f, 0.f, 0.f, 0.f, 0.f, 0.f, 0.f, 0.f};
    const float* xrow = x + (size_t)(m0 + l15) * DIM;

    #pragma unroll 4
    for (int k = 0; k < DIM; k += 4) {
        const int ka = k + lhalf * 2;
        v2f a, bm;
        // A 16x4: lanes 0-15 hold K={0,1}, lanes 16-31 hold K={2,3} of this step
        a.x = xrow[ka];
        a.y = xrow[ka + 1];
        // B 4x16: row K striped across lanes (N = l15), same K split by lane half
        bm.x = W[(size_t)ka       * DIM + n0 + l15];
        bm.y = W[(size_t)(ka + 1) * DIM + n0 + l15];
        c = __builtin_amdgcn_wmma_f32_16x16x4_f32(false, a, false, bm,
                                                  (short)0, c, false, false);
    }
    // D 16x16 f32: VGPR i -> M = i (lanes 0-15) / M = i+8 (lanes 16-31), N = l15
    #pragma unroll
    for (int i = 0; i < 8; ++i) {
        const int m   = m0 + i + lhalf * 8;
        const int col = n0 + l15;
        xl[(size_t)m * DIM + col] = c[i] + b[col];
    }
}

// ---------------- K2: per-edge attention logits + segment max ----------------
// One wave per edge; lane owns 4 channels (head h = lane>>3).
__global__ void k2_edge_alpha(const float* __restrict__ xl,
                              const long long* __restrict__ ei,   // [2,E] int64
                              const float* __restrict__ eattr,    // [E,1]
                              const float* __restrict__ We,       // [1,128]
                              const float* __restrict__ att,      // [4,32] flat 128
                              float* __restrict__ alpha,          // [E,4]
                              float* __restrict__ amax) {         // [N,4]
    const int e    = (blockIdx.x * blockDim.x + threadIdx.x) >> 5;
    const int lane = threadIdx.x & 31;
    if (e >= N_EDGES) return;

    const long long s = ei[e];
    const long long d = ei[N_EDGES + e];
    const float ev    = eattr[e];

    const float4 xs = reinterpret_cast<const float4*>(xl + (size_t)s * DIM)[lane];
    const float4 xd = reinterpret_cast<const float4*>(xl + (size_t)d * DIM)[lane];
    const float4 w  = reinterpret_cast<const float4*>(We)[lane];
    const float4 at = reinterpret_cast<const float4*>(att)[lane];

    float m0 = xs.x + xd.x + ev * w.x;
    float m1 = xs.y + xd.y + ev * w.y;
    float m2 = xs.z + xd.z + ev * w.z;
    float m3 = xs.w + xd.w + ev * w.w;
    m0 = (m0 > 0.f) ? m0 : NEG_SLOPE * m0;
    m1 = (m1 > 0.f) ? m1 : NEG_SLOPE * m1;
    m2 = (m2 > 0.f) ? m2 : NEG_SLOPE * m2;
    m3 = (m3 > 0.f) ? m3 : NEG_SLOPE * m3;

    float partial = m0 * at.x + m1 * at.y + m2 * at.z + m3 * at.w;
    // reduce over the 8 lanes of this head (xor offsets stay in the group)
    partial += __shfl_xor(partial, 1, 32);
    partial += __shfl_xor(partial, 2, 32);
    partial += __shfl_xor(partial, 4, 32);

    if ((lane & 7) == 0) {
        const int h = lane >> 3;
        alpha[(size_t)e * HEADS + h] = partial;
        atomicMax(&amax[(size_t)d * HEADS + h], partial);   // native f32 max atomic
    }
}

// ---------------- K3: ea = exp(alpha - amax[dst]); denom += ea ----------------
__global__ void k3_softmax_denom(const long long* __restrict__ ei,
                                 float* __restrict__ alpha,    // in: logits, out: ea
                                 const float* __restrict__ amax,
                                 float* __restrict__ denom) {
    const int idx = blockIdx.x * blockDim.x + threadIdx.x;     // over E*HEADS
    if (idx >= N_EDGES * HEADS) return;
    const int e = idx >> 2;
    const int h = idx & 3;
    const long long d = ei[N_EDGES + e];
    const float v = expf(alpha[idx] - amax[(size_t)d * HEADS + h]);
    alpha[idx] = v;
    atomicAdd(&denom[(size_t)d * HEADS + h], v);
}

// ---------------- K4: out[dst] += (ea/denom) * xl[src] ----------------
__global__ void k4_scatter(const float* __restrict__ xl,
                           const long long* __restrict__ ei,
                           const float* __restrict__ alpha,   // holds ea
                           const float* __restrict__ denom,
                           float* __restrict__ outacc) {
    const int e    = (blockIdx.x * blockDim.x + threadIdx.x) >> 5;
    const int lane = threadIdx.x & 31;
    if (e >= N_EDGES) return;

    const long long s = ei[e];
    const long long d = ei[N_EDGES + e];
    const int h = lane >> 3;

    const float w = alpha[(size_t)e * HEADS + h] /
                    (denom[(size_t)d * HEADS + h] + SM_EPS);

    const float4 xs = reinterpret_cast<const float4*>(xl + (size_t)s * DIM)[lane];
    float* o = outacc + (size_t)d * DIM + lane * 4;
    atomicAdd(o + 0, w * xs.x);
    atomicAdd(o + 1, w * xs.y);
    atomicAdd(o + 2, w * xs.z);
    atomicAdd(o + 3, w * xs.w);
}

// ---------------- K5: bias + LayerNorm + exact GELU + residual ----------------
// One wave per node; lane owns 4 channels.
__global__ void k5_norm_gelu(const float* __restrict__ x,
                             const float* __restrict__ outacc,
                             const float* __restrict__ bias,
                             const float* __restrict__ gamma,
                             const float* __restrict__ beta,
                             float* __restrict__ out) {
    const int n    = (blockIdx.x * blockDim.x + threadIdx.x) >> 5;
    const int lane = threadIdx.x & 31;
    if (n >= N_NODES) return;

    const float4 o  = reinterpret_cast<const float4*>(outacc + (size_t)n * DIM)[lane];
    const float4 bb = reinterpret_cast<const float4*>(bias)[lane];
    float h0 = o.x + bb.x, h1 = o.y + bb.y, h2 = o.z + bb.z, h3 = o.w + bb.w;

    float sum = h0 + h1 + h2 + h3;
    float sq  = h0 * h0 + h1 * h1 + h2 * h2 + h3 * h3;
    #pragma unroll
    for (int off = 1; off < 32; off <<= 1) {
        sum += __shfl_xor(sum, off, 32);
        sq  += __shfl_xor(sq,  off, 32);
    }
    const float mean = sum * (1.0f / DIM);
    const float var  = sq * (1.0f / DIM) - mean * mean;
    const float inv  = rsqrtf(var + LN_EPS);

    const float4 g  = reinterpret_cast<const float4*>(gamma)[lane];
    const float4 be = reinterpret_cast<const float4*>(beta)[lane];
    const float4 xr = reinterpret_cast<const float4*>(x + (size_t)n * DIM)[lane];

    const float k = 0.70710678118654752f;  // 1/sqrt(2)
    float n0 = (h0 - mean) * inv * g.x + be.x;
    float n1 = (h1 - mean) * inv * g.y + be.y;
    float n2 = (h2 - mean) * inv * g.z + be.z;
    float n3 = (h3 - mean) * inv * g.w + be.w;
    float4 r;
    r.x = xr.x + 0.5f * n0 * (1.0f + erff(n0 * k));
    r.y = xr.y + 0.5f * n1 * (1.0f + erff(n1 * k));
    r.z = xr.z + 0.5f * n2 * (1.0f + erff(n2 * k));
    r.w = xr.w + 0.5f * n3 * (1.0f + erff(n3 * k));
    reinterpret_cast<float4*>(out + (size_t)n * DIM)[lane] = r;
}

extern "C" void kernel_launch(void* const* d_in, const int* in_sizes, int n_in,
                              void* d_out, int out_size, void* d_ws, size_t ws_size,
                              hipStream_t stream) {
    const float*     x      = (const float*)d_in[0];
    const long long* ei     = (const long long*)d_in[1];   // int64 [2,E]
    const float*     eattr  = (const float*)d_in[2];
    const float*     W_l    = (const float*)d_in[3];
    const float*     b_l    = (const float*)d_in[4];
    const float*     W_e    = (const float*)d_in[5];
    const float*     att    = (const float*)d_in[6];
    const float*     bias   = (const float*)d_in[7];
    const float*     gamma  = (const float*)d_in[8];
    const float*     beta   = (const float*)d_in[9];
    float*           out    = (float*)d_out;
    float*           ws     = (float*)d_ws;

    float* xl     = ws + OFF_XL;
    float* alpha  = ws + OFF_ALPHA;
    float* amax   = ws + OFF_AMAX;
    float* denom  = ws + OFF_DENOM;
    float* outacc = ws + OFF_OUT;

    // K0: init accumulators (must run every call; harness doesn't re-poison)
    k0_init<<<(N_NODES * DIM + 255) / 256, 256, 0, stream>>>(ws);

    // K1: WMMA GEMM, 3125 row tiles x (8 waves = 8 col tiles)
    k1_gemm_xl<<<N_NODES / 16, 256, 0, stream>>>(x, W_l, b_l, xl);

    // K2: one wave per edge -> logits + segment max
    k2_edge_alpha<<<(N_EDGES * 32 + 255) / 256, 256, 0, stream>>>(
        xl, ei, eattr, W_e, att, alpha, amax);

    // K3: exp + denom
    k3_softmax_denom<<<(N_EDGES * HEADS + 255) / 256, 256, 0, stream>>>(
        ei, alpha, amax, denom);

    // K4: one wave per edge -> weighted scatter-add (L2-resident atomics)
    k4_scatter<<<(N_EDGES * 32 + 255) / 256, 256, 0, stream>>>(
        xl, ei, alpha, denom, outacc);

    // K5: one wave per node -> bias + LN + exact GELU + residual
    k5_norm_gelu<<<(N_NODES * 32 + 255) / 256, 256, 0, stream>>>(
        x, outacc, bias, gamma, beta, out);
}